// GCN_17308718202961
// MI455X (gfx1250) — compile-verified
//
#include <hip/hip_runtime.h>
#include <hip/hip_bf16.h>

typedef __attribute__((ext_vector_type(2))) float v2f;
typedef __attribute__((ext_vector_type(8))) float v8f;

#define TB 256

// ---------------- degree / normalization ----------------

__global__ void k_init_deg(float* __restrict__ deg, int n) {
  int i = blockIdx.x * blockDim.x + threadIdx.x;
  if (i < n) deg[i] = 1.0f;  // self-loop contributes 1
}

__global__ void k_accum_deg(const int* __restrict__ dst, float* __restrict__ deg, int E) {
  int e = blockIdx.x * blockDim.x + threadIdx.x;
  if (e < E) atomicAdd(&deg[dst[e]], 1.0f);
}

__global__ void k_make_dinv(float* __restrict__ deg, int n) {
  int i = blockIdx.x * blockDim.x + threadIdx.x;
  if (i < n) deg[i] = rsqrtf(deg[i]);  // deg >= 1 always (self-loops)
}

// ---------------- fp32 WMMA GEMM: C[M,N] = op(A)[M,K] * B[K,N] ----------------
// Block = 8 waves; block covers 8 consecutive 16-row M-tiles x one 64-col group.
// B strip staged in LDS (transposed, +4 float pad -> conflict-free ds_load_b64),
// in 128-deep K chunks. Each wave: 16x64 strip via V_WMMA_F32_16X16X4_F32.
// fp32 fragment layout (ISA 7.12.2): A 16x4: lanes 0-15 -> row=lane, K={k,k+1};
// lanes 16-31 -> row=lane-16, K={k+2,k+3}. B 4x16 symmetric (lane = column).
// C/D: vgpr j, lanes 0-15 -> (M=j, N=lane); lanes 16-31 -> (M=j+8, N=lane-16).
template<int K, int N, bool RELU>
__global__ void k_gemm_wmma(const float* __restrict__ A, const float* __restrict__ B,
                            float* __restrict__ C, int M) {
  constexpr int NT   = 4;
  constexpr int COLS = 16 * NT;              // 64 columns per block
  constexpr int KC   = 128;                  // K chunk staged in LDS
  constexpr int LDK  = KC + 4;               // padded row stride (floats)
  static_assert(K % KC == 0, "K must be a multiple of KC");
  static_assert(N % COLS == 0, "N must be a multiple of 64");
  __shared__ float Bs[COLS * LDK];           // 64*132*4 = 33 KB

  constexpr int ngroups = N / COLS;
  int bm = blockIdx.x / ngroups;
  int ng = blockIdx.x % ngroups;
  int n0 = ng * COLS;

  int wid  = threadIdx.x >> 5;
  int lane = threadIdx.x & 31;
  int half = lane >> 4;                      // K pair {0,1} vs {2,3}
  int l    = lane & 15;                      // row (A) / column (B)

  int mt     = bm * 8 + wid;
  int mtiles = M >> 4;
  bool active = (mt < mtiles);
  int m0 = mt << 4;

  v8f acc[NT] = {};
  const float* Arow = active ? (A + (size_t)(m0 + l) * K + half * 2) : A;
  const float* Bl   = Bs + l * LDK + half * 2;

  for (int kc = 0; kc < K; kc += KC) {
    __syncthreads();  // previous chunk fully consumed
    // cooperative stage: Bs[c][k] = B[kc+k][n0+c]  (global reads contiguous in c)
    for (int idx = threadIdx.x; idx < KC * COLS; idx += TB) {
      int k = idx / COLS;
      int c = idx % COLS;
      Bs[c * LDK + k] = B[(size_t)(kc + k) * N + n0 + c];
    }
    __syncthreads();

    if (active) {
#pragma unroll 4
      for (int k = 0; k < KC; k += 4) {
        float a0 = Arow[kc + k];
        float a1 = Arow[kc + k + 1];
        if (RELU) { a0 = fmaxf(a0, 0.0f); a1 = fmaxf(a1, 0.0f); }
        v2f a; a.x = a0; a.y = a1;
#pragma unroll
        for (int t = 0; t < NT; ++t) {
          const float* bp = Bl + t * 16 * LDK + k;   // single aligned ds_load_b64
          v2f b; b.x = bp[0]; b.y = bp[1];
          acc[t] = __builtin_amdgcn_wmma_f32_16x16x4_f32(
              false, a, false, b, (short)0, acc[t], false, false);
        }
      }
    }
  }

  if (active) {
    int rbase = m0 + half * 8;
#pragma unroll
    for (int t = 0; t < NT; ++t) {
#pragma unroll
      for (int j = 0; j < 8; ++j) {
        C[(size_t)(rbase + j) * N + n0 + t * 16 + l] = acc[t][j];
      }
    }
  }
}

// ---------------- aggregation ----------------

// AGG[i,f] = H[i,f]*dinv[i]^2 + b[f]   (self-loop message + bias)
__global__ void k_init_agg(const float* __restrict__ H, const float* __restrict__ dinv,
                           const float* __restrict__ b, float* __restrict__ AGG,
                           int n, int F) {
  long long i = (long long)blockIdx.x * blockDim.x + threadIdx.x;
  long long total = (long long)n * F;
  if (i >= total) return;
  int row = (int)(i / F);
  int col = (int)(i % F);
  float di = dinv[row];
  AGG[i] = H[i] * di * di + b[col];
}

// Edge scatter: AGG[dst] += H[src] * dinv[src]*dinv[dst].
// F/4 lanes per edge, float4 gathers (contiguous per wave), fp32 atomic scatter.
template<int F>
__global__ void k_edge_scatter(const float* __restrict__ H, float* __restrict__ AGG,
                               const int* __restrict__ src, const int* __restrict__ dst,
                               const float* __restrict__ dinv, int E) {
  constexpr int LPE = F / 4;       // lanes per edge
  constexpr int EPW = 32 / LPE;    // edges per wave
  long long gwave = (long long)((blockIdx.x * blockDim.x + threadIdx.x) >> 5);
  int lane = threadIdx.x & 31;
  int sub  = lane / LPE;
  int fl   = lane % LPE;
  long long e = gwave * EPW + sub;
  if (e >= E) return;

  int s = src[e];
  int d = dst[e];
  float norm = dinv[s] * dinv[d];

  const float4* hp = (const float4*)(H + (size_t)s * F);
  float4 v = hp[fl];
  float* ap = AGG + (size_t)d * F + fl * 4;
  atomicAdd(ap + 0, v.x * norm);
  atomicAdd(ap + 1, v.y * norm);
  atomicAdd(ap + 2, v.z * norm);
  atomicAdd(ap + 3, v.w * norm);
}

// ---------------- log_softmax over 64 cols, z = x / T ----------------

__global__ void k_log_softmax64(float* __restrict__ out, int n, float invT) {
  int wave = (blockIdx.x * blockDim.x + threadIdx.x) >> 5;
  int lane = threadIdx.x & 31;
  if (wave >= n) return;
  float* row = out + (size_t)wave * 64;
  float z0 = row[lane]      * invT;
  float z1 = row[lane + 32] * invT;
  float m = fmaxf(z0, z1);
#pragma unroll
  for (int off = 16; off > 0; off >>= 1) m = fmaxf(m, __shfl_xor(m, off, 32));
  float s = expf(z0 - m) + expf(z1 - m);
#pragma unroll
  for (int off = 16; off > 0; off >>= 1) s += __shfl_xor(s, off, 32);
  float lg = m + logf(s);
  row[lane]      = z0 - lg;
  row[lane + 32] = z1 - lg;
}

// ---------------- launcher ----------------

extern "C" void kernel_launch(void* const* d_in, const int* in_sizes, int n_in,
                              void* d_out, int out_size, void* d_ws, size_t ws_size,
                              hipStream_t stream) {
  const float* x  = (const float*)d_in[0];  // [n, 256]
  const int*   ei = (const int*)d_in[1];    // [2, E]
  const float* W1 = (const float*)d_in[2];  // [256, 128]
  const float* b1 = (const float*)d_in[3];  // [128]
  const float* W2 = (const float*)d_in[4];  // [128, 64]
  const float* b2 = (const float*)d_in[5];  // [64]

  const int IN = 256, HID = 128, OUT = 64;
  int n = in_sizes[0] / IN;   // 50000
  int E = in_sizes[1] / 2;    // 1600000
  const int* src = ei;        // row 0
  const int* dst = ei + E;    // row 1

  // workspace carve-up
  float* ws = (float*)d_ws;
  size_t o = 0;
  float* deg  = ws + o; o += ((size_t)n + 255) & ~(size_t)255;  // becomes dinv in place
  float* H1   = ws + o; o += (size_t)n * HID;
  float* AGG1 = ws + o; o += (size_t)n * HID;
  float* H2   = ws + o; o += (size_t)n * OUT;
  float* AGG2 = (float*)d_out;                                  // [n, 64]

  int mtiles  = n / 16;                 // 3125
  int mblocks = (mtiles + 7) / 8;       // 391

  // 1. degree + dinv
  k_init_deg <<<(n + TB - 1) / TB, TB, 0, stream>>>(deg, n);
  k_accum_deg<<<(E + TB - 1) / TB, TB, 0, stream>>>(dst, deg, E);
  k_make_dinv<<<(n + TB - 1) / TB, TB, 0, stream>>>(deg, n);

  // 2. H1 = X @ W1   (fp32 WMMA, LDS-staged B)
  k_gemm_wmma<256, 128, false><<<mblocks * (HID / 64), TB, 0, stream>>>(x, W1, H1, n);

  // 3. AGG1 = self-loop + bias, then edge scatter
  {
    long long tot = (long long)n * HID;
    k_init_agg<<<(unsigned)((tot + TB - 1) / TB), TB, 0, stream>>>(H1, deg, b1, AGG1, n, HID);
    long long waves = E;  // 1 edge per wave at F=128
    k_edge_scatter<128><<<(unsigned)((waves * 32 + TB - 1) / TB), TB, 0, stream>>>(
        H1, AGG1, src, dst, deg, E);
  }

  // 4. H2 = relu(AGG1) @ W2   (ReLU fused into A load)
  k_gemm_wmma<128, 64, true><<<mblocks * (OUT / 64), TB, 0, stream>>>(AGG1, W2, H2, n);

  // 5. AGG2 (in d_out) = self-loop + bias, then edge scatter (2 edges/wave at F=64)
  {
    long long tot = (long long)n * OUT;
    k_init_agg<<<(unsigned)((tot + TB - 1) / TB), TB, 0, stream>>>(H2, deg, b2, AGG2, n, OUT);
    long long waves = ((long long)E + 1) / 2;
    k_edge_scatter<64><<<(unsigned)((waves * 32 + TB - 1) / TB), TB, 0, stream>>>(
        H2, AGG2, src, dst, deg, E);
  }

  // 6. log_softmax(AGG2 / 0.2) in place, one wave per row
  k_log_softmax64<<<(n * 32 + TB - 1) / TB, TB, 0, stream>>>(AGG2, n, 5.0f);
}